// ESMGridSample_88467736363595
// MI455X (gfx1250) — compile-verified
//
#include <hip/hip_runtime.h>
#include <hip/hip_bf16.h>

// Bilinear grid sample (zeros padding, align_corners=False) for
//   img:  [16, 1, 1024, 1024] f32
//   grid: [16, 1024, 1024, 2] f32  (x, y) in [-1, 1]
//   out:  [16, 1, 1024, 1024] f32
//
// Memory-bound kernel. MI455X strategy:
//  * 64MB image working set fits in the 192MB global L2 -> keep image loads
//    at default (RT) temporal policy so gathers hit L2.
//  * grid (128MB) and output (64MB) are streamed exactly once -> NT hints so
//    they do not evict the image from L2.
//  * global_prefetch_b8 on the grid stream (CDNA5 prefetch path).

typedef float v2f __attribute__((ext_vector_type(2)));

#ifndef GS_BLOCK
#define GS_BLOCK 256   // 8 wave32 waves per block
#endif

__global__ __launch_bounds__(GS_BLOCK) void ESMGridSample_kernel(
    const float* __restrict__ img,    // [N*H*W], C == 1
    const float* __restrict__ grid,   // [N*H*W*2]
    float* __restrict__ out,          // [N*H*W]
    int total)                        // N*H*W
{
    constexpr int W    = 1024;
    constexpr int H    = 1024;
    constexpr int LOGW = 10;
    constexpr int LOGHW = 20;         // H*W = 2^20

    int i = blockIdx.x * GS_BLOCK + threadIdx.x;
    if (i >= total) return;

    // Prefetch the grid stream ~128KB ahead (clamped in-range; speculative,
    // no VGPR return, no counter). Emits global_prefetch_b8 on gfx1250.
    int pf = i + 16384;
    if (pf >= total) pf = total - 1;
    __builtin_prefetch((const v2f*)grid + pf, /*rw=*/0, /*loc=*/0);

    // Streaming, read-once grid coordinate: non-temporal 8B load.
    v2f g = __builtin_nontemporal_load((const v2f*)grid + i);

    // ix = (gx + 1) * (W/2) - 0.5 = gx*512 + 511.5
    float ix = fmaf(g.x, 512.0f, 511.5f);
    float iy = fmaf(g.y, 512.0f, 511.5f);

    float x0f = floorf(ix);
    float y0f = floorf(iy);
    float wx  = ix - x0f;
    float wy  = iy - y0f;

    int x0 = (int)x0f;
    int y0 = (int)y0f;
    int x1 = x0 + 1;
    int y1 = y0 + 1;

    // Per-corner validity (zeros padding), then clamp so loads are in-bounds.
    bool vx0 = (unsigned)x0 < (unsigned)W;
    bool vx1 = (unsigned)x1 < (unsigned)W;
    bool vy0 = (unsigned)y0 < (unsigned)H;
    bool vy1 = (unsigned)y1 < (unsigned)H;

    int xc0 = min(max(x0, 0), W - 1);
    int xc1 = min(max(x1, 0), W - 1);
    int yc0 = min(max(y0, 0), H - 1);
    int yc1 = min(max(y1, 0), H - 1);

    // Image slice for batch n = i >> 20 (C == 1).
    const float* __restrict__ imn = img + ((i >> LOGHW) << LOGHW);

    int r0 = yc0 << LOGW;
    int r1 = yc1 << LOGW;

    // Gathers: default RT policy -> resident in 192MB L2 (whole 64MB image
    // set fits), so the random accesses are L2 hits after cold fill.
    float v00 = imn[r0 + xc0];
    float v01 = imn[r0 + xc1];
    float v10 = imn[r1 + xc0];
    float v11 = imn[r1 + xc1];

    // Zero out out-of-bounds corners (branchless selects).
    v00 = (vx0 && vy0) ? v00 : 0.0f;
    v01 = (vx1 && vy0) ? v01 : 0.0f;
    v10 = (vx0 && vy1) ? v10 : 0.0f;
    v11 = (vx1 && vy1) ? v11 : 0.0f;

    float owx = 1.0f - wx;
    float owy = 1.0f - wy;
    float top = fmaf(v01, wx, v00 * owx);
    float bot = fmaf(v11, wx, v10 * owx);
    float r   = fmaf(bot, wy, top * owy);

    // Write-once streaming output: non-temporal store.
    __builtin_nontemporal_store(r, out + i);
}

extern "C" void kernel_launch(void* const* d_in, const int* in_sizes, int n_in,
                              void* d_out, int out_size, void* d_ws, size_t ws_size,
                              hipStream_t stream) {
    (void)in_sizes; (void)n_in; (void)d_ws; (void)ws_size;

    const float* img  = (const float*)d_in[0];   // source_depth [16,1,1024,1024]
    const float* grid = (const float*)d_in[1];   // pr           [16,1024,1024,2]
    // d_in[2] (gt_map) is unused by the reference.
    float* out = (float*)d_out;                  // [16,1,1024,1024]

    int total = out_size;                        // 16 * 1024 * 1024
    int blocks = (total + GS_BLOCK - 1) / GS_BLOCK;
    ESMGridSample_kernel<<<blocks, GS_BLOCK, 0, stream>>>(img, grid, out, total);
}